// KeyValueMemoryModule_88261577933101
// MI455X (gfx1250) — compile-verified
//
#include <hip/hip_runtime.h>
#include <math.h>

typedef float v2f __attribute__((ext_vector_type(2)));
typedef float v4f __attribute__((ext_vector_type(4)));
typedef float v8f __attribute__((ext_vector_type(8)));

constexpr int BATCH = 512;
constexpr int MROWS = 2048;
constexpr int FD    = 64;     // KD == VD == 64
constexpr int NW    = 8;      // waves per block (256 threads, wave32)
constexpr int TS    = 68;     // padded LDS tile row stride (floats) -> conflict-free
constexpr int NT    = MROWS / 16;   // 128 tiles of 16 rows

__global__ __launch_bounds__(256)
void kv_mem_kernel(const float* __restrict__ new_key,
                   const float* __restrict__ new_value,
                   const float* __restrict__ key_mem,
                   const float* __restrict__ val_mem,
                   const float* __restrict__ gate,
                   const float* __restrict__ conf_w,
                   const float* __restrict__ conf_b,
                   const int*   __restrict__ iteration,
                   float* __restrict__ out_km,
                   float* __restrict__ out_vm,
                   float* __restrict__ out_rd)
{
    __shared__ float s_sim[MROWS];                 // sim, then softmax numerators
    __shared__ __align__(16) float s_nv[FD];
    __shared__ __align__(16) float s_nk[FD];
    __shared__ float s_tile[NW][16 * TS];          // per-wave staging tiles
    __shared__ float s_red[3 * NW];                // max / sum / conf-dot partials
    __shared__ float s_rp[NW * FD];                // per-wave read partials

    const int b    = blockIdx.x;
    const int tid  = threadIdx.x;
    const int wave = tid >> 5;
    const int lane = tid & 31;
    const int lo   = lane & 15;     // N / row-within-16 index
    const int hi   = lane >> 4;     // lane-half: selects K slots {0,1} vs {2,3}

    const int   it = iteration[b];
    const float cw = conf_w[0];
    const float cb = conf_b[0];

    if (tid < FD)            s_nv[tid]      = new_value[(size_t)b * FD + tid];
    else if (tid < 2 * FD)   s_nk[tid - FD] = new_key  [(size_t)b * FD + (tid - FD)];
    __syncthreads();

    const size_t base = (size_t)b * MROWS * FD;

    // ---------------- Phase 1: sim = (VM * mask) . new_value  +  VM copy ----------------
    {
        const v4f* src = (const v4f*)(val_mem + base);
        v4f*       dst = (v4f*)(out_vm + base);
        float* tile = &s_tile[wave][0];

        for (int t = wave; t < NT; t += NW) {
            const int r0 = t * 16;
            #pragma unroll
            for (int i = 0; i < 8; ++i) {
                const int flat = i * 32 + lane;        // 0..255
                const int rr   = flat >> 4;            // tile row 0..15
                const int c4   = flat & 15;            // float4 column 0..15
                const int gr   = r0 + rr;              // global row
                v4f v = __builtin_nontemporal_load(&src[(size_t)gr * 16 + c4]);
                *(v4f*)&tile[rr * TS + c4 * 4] = v;
                v4f o;
                if (gr == it)              o = ((const v4f*)s_nv)[c4];
                else if (gr == MROWS - 1)  o = (v4f){0.f, 0.f, 0.f, 0.f};
                else                       o = v;
                __builtin_nontemporal_store(o, &dst[(size_t)gr * 16 + c4]);
            }
            asm volatile("" ::: "memory");   // keep WMMA LDS reads after staging stores

            const float msk = ((r0 + lo) <= it) ? 1.0f : 0.0f;
            v8f c = {0.f, 0.f, 0.f, 0.f, 0.f, 0.f, 0.f, 0.f};
            #pragma unroll
            for (int k0 = 0; k0 < FD; k0 += 4) {
                const int kk = k0 + 2 * hi;            // K slot per ISA A/B layout
                v2f a, bv;
                a.x  = tile[lo * TS + kk]     * msk;   // A[M=lo][K=kk]
                a.y  = tile[lo * TS + kk + 1] * msk;   // A[M=lo][K=kk+1]
                bv.x = s_nv[kk];                       // B[K=kk][*]  (broadcast cols)
                bv.y = s_nv[kk + 1];                   // B[K=kk+1][*]
                c = __builtin_amdgcn_wmma_f32_16x16x4_f32(
                        false, a, false, bv, (short)0, c, false, false);
            }
            // D: vgpr j holds row (j + 8*hi); all 16 columns equal -> 2 lanes write
            if (lo == 0) {
                #pragma unroll
                for (int j = 0; j < 8; ++j) s_sim[r0 + hi * 8 + j] = c[j];
            }
            asm volatile("" ::: "memory");
        }
    }
    __syncthreads();

    // ---------------- Phase 2: softmax + conf dot ----------------
    float mx = -3.4e38f;
    for (int i = tid; i < MROWS; i += 256) mx = fmaxf(mx, s_sim[i]);
    #pragma unroll
    for (int off = 16; off; off >>= 1) mx = fmaxf(mx, __shfl_xor(mx, off, 32));
    if (lane == 0) s_red[wave] = mx;
    __syncthreads();
    mx = s_red[0];
    #pragma unroll
    for (int w = 1; w < NW; ++w) mx = fmaxf(mx, s_red[w]);

    float sum = 0.f, cnum = 0.f;
    for (int i = tid; i < MROWS; i += 256) {
        const float s  = s_sim[i];
        const float e  = expf(s - mx);
        const float cf = 1.f / (1.f + expf(-(s * cw + cb)));   // conf = sigmoid(sim*w+b)
        sum  += e;
        cnum += e * cf;
        s_sim[i] = e;                                          // numerator
    }
    #pragma unroll
    for (int off = 16; off; off >>= 1) {
        sum  += __shfl_xor(sum,  off, 32);
        cnum += __shfl_xor(cnum, off, 32);
    }
    if (lane == 0) { s_red[NW + wave] = sum; s_red[2 * NW + wave] = cnum; }
    __syncthreads();
    sum = 0.f; cnum = 0.f;
    #pragma unroll
    for (int w = 0; w < NW; ++w) { sum += s_red[NW + w]; cnum += s_red[2 * NW + w]; }
    const float inv       = 1.f / sum;
    const float read_conf = cnum * inv;    // sum_n wv[n] * conf[n]

    // ---------------- Phase 3: read[d] = sum_n wv[n]*KM[n,d]  +  KM copy ----------------
    {
        const v4f* src = (const v4f*)(key_mem + base);
        v4f*       dst = (v4f*)(out_km + base);
        float* tile = &s_tile[wave][0];
        v8f acc[4];
        #pragma unroll
        for (int ct = 0; ct < 4; ++ct) acc[ct] = (v8f){0.f,0.f,0.f,0.f,0.f,0.f,0.f,0.f};

        for (int t = wave; t < NT; t += NW) {
            const int r0 = t * 16;
            #pragma unroll
            for (int i = 0; i < 8; ++i) {
                const int flat = i * 32 + lane;
                const int rr   = flat >> 4;
                const int c4   = flat & 15;
                const int gr   = r0 + rr;
                v4f v = __builtin_nontemporal_load(&src[(size_t)gr * 16 + c4]);
                *(v4f*)&tile[rr * TS + c4 * 4] = v;
                v4f o;
                if (gr == it)              o = ((const v4f*)s_nk)[c4];
                else if (gr == MROWS - 1)  o = (v4f){0.f, 0.f, 0.f, 0.f};
                else                       o = v;
                __builtin_nontemporal_store(o, &dst[(size_t)gr * 16 + c4]);
            }
            asm volatile("" ::: "memory");

            #pragma unroll
            for (int kc = 0; kc < 4; ++kc) {            // 4 K-chunks of 4 rows
                const int kb    = r0 + kc * 4 + 2 * hi;
                const int rbase = (kc * 4 + 2 * hi) * TS;
                v2f a;
                a.x = s_sim[kb]     * inv;              // A[M=*][K] = wv (broadcast rows)
                a.y = s_sim[kb + 1] * inv;
                #pragma unroll
                for (int ct = 0; ct < 4; ++ct) {        // 4 column tiles -> 64 features
                    v2f bv;
                    bv.x = tile[rbase      + ct * 16 + lo];   // B[K][N=lo] = KM[row][d]
                    bv.y = tile[rbase + TS + ct * 16 + lo];
                    acc[ct] = __builtin_amdgcn_wmma_f32_16x16x4_f32(
                                  false, a, false, bv, (short)0, acc[ct], false, false);
                }
            }
            asm volatile("" ::: "memory");
        }
        // every D row equal (A broadcast over M); lanes 0..15, vgpr0 -> read[d0+lo]
        if (hi == 0) {
            #pragma unroll
            for (int ct = 0; ct < 4; ++ct)
                s_rp[wave * FD + ct * 16 + lo] = acc[ct][0];
        }
    }
    __syncthreads();

    // ---------------- Final: gate, not_first, write read output ----------------
    if (tid < FD + 1) {
        float r;
        if (tid < FD) {
            r = 0.f;
            #pragma unroll
            for (int w = 0; w < NW; ++w) r += s_rp[w * FD + tid];
        } else {
            r = read_conf;
        }
        const float g = gate[(size_t)b * (FD + 1) + tid];
        r *= 1.f / (1.f + expf(-g));
        if (it <= 1) r = 0.f;      // not_first == (it > 1)
        out_rd[(size_t)b * (FD + 1) + tid] = r;
    }
}

extern "C" void kernel_launch(void* const* d_in, const int* in_sizes, int n_in,
                              void* d_out, int out_size, void* d_ws, size_t ws_size,
                              hipStream_t stream) {
    (void)in_sizes; (void)n_in; (void)out_size; (void)d_ws; (void)ws_size;
    const float* new_key   = (const float*)d_in[0];
    const float* new_value = (const float*)d_in[1];
    const float* key_mem   = (const float*)d_in[2];
    const float* val_mem   = (const float*)d_in[3];
    const float* gate      = (const float*)d_in[4];
    const float* conf_w    = (const float*)d_in[5];
    const float* conf_b    = (const float*)d_in[6];
    const int*   iter      = (const int*)d_in[7];

    float* out    = (float*)d_out;
    float* out_km = out;                                     // (B, M, KD)
    float* out_vm = out + (size_t)BATCH * MROWS * FD;        // (B, M, VD)
    float* out_rd = out + 2 * (size_t)BATCH * MROWS * FD;    // (B, KD+1)

    kv_mem_kernel<<<BATCH, 256, 0, stream>>>(new_key, new_value, key_mem, val_mem,
                                             gate, conf_w, conf_b, iter,
                                             out_km, out_vm, out_rd);
}